// Actor2ActorAttention_52140902974096
// MI455X (gfx1250) — compile-verified
//
#include <hip/hip_runtime.h>
#include <hip/hip_bf16.h>

typedef float v2f __attribute__((ext_vector_type(2)));
typedef float v8f __attribute__((ext_vector_type(8)));

#define D_FEAT 128
#define EDGE_IN 258   // 128 src + 2 rel + 128 dst
#define PCOLS   256   // [P_src | P_dst]

// ---- monotone float<->uint encoding for order-invariant atomic max --------
__device__ __forceinline__ unsigned enc_f32(float f) {
  unsigned u = __float_as_uint(f);
  return (u & 0x80000000u) ? ~u : (u | 0x80000000u);
}
__device__ __forceinline__ float dec_f32(unsigned u) {
  return __uint_as_float((u & 0x80000000u) ? (u ^ 0x80000000u) : ~u);
}
__device__ __forceinline__ void atomAddF(float* p, float v) {
  unsafeAtomicAdd(p, v);   // hardware global_atomic_add_f32
}

// ---- per-layer init of reduction buffers ----------------------------------
__global__ void a2a_init_kernel(float* __restrict__ agg, float* __restrict__ denom,
                                unsigned* __restrict__ maxu, int N) {
  int i = blockIdx.x * blockDim.x + threadIdx.x;
  if (i < N * D_FEAT) agg[i] = 0.f;
  if (i < N) { denom[i] = 0.f; maxu[i] = 0u; }   // 0 encodes < any real value
}

// ---- extract the two rel-columns of W_emb (strided by 258) ----------------
__global__ void a2a_wrel_kernel(const float* __restrict__ We, float* __restrict__ wrel) {
  int d = threadIdx.x;                 // 128 threads
  wrel[d]           = We[(size_t)d * EDGE_IN + D_FEAT];
  wrel[D_FEAT + d]  = We[(size_t)d * EDGE_IN + D_FEAT + 1];
}

// ---- P = feat @ [W_src^T | W_dst^T]  via V_WMMA_F32_16X16X4_F32 ----------
// One wave computes a 16-row x 64-col tile of P (4 x v8f accumulators).
__global__ __launch_bounds__(256) void a2a_proj_wmma_kernel(
    const float* __restrict__ feat, const float* __restrict__ We,
    float* __restrict__ P, int N) {
  int wid  = blockIdx.x * 8 + (threadIdx.x >> 5);
  int lane = threadIdx.x & 31;
  int rowtiles = (N + 15) >> 4;
  if (wid >= rowtiles * 4) return;      // whole-wave uniform exit (EXEC stays full)
  int rowtile = wid >> 2;
  int colgrp  = wid & 3;                // 64 output columns per wave
  int m0   = rowtile << 4;
  int half = lane >> 4;                 // lane half selects K pair / M+8
  int mr   = lane & 15;

  int arow = m0 + mr; if (arow > N - 1) arow = N - 1;   // clamp, no divergence
  const float* aptr = feat + (size_t)arow * D_FEAT + 2 * half;

  v8f acc[4] = {};
  for (int k0 = 0; k0 < D_FEAT; k0 += 4) {
    // A 16x4 f32: v0 = A[m, k0+2*half], v1 = A[m, k0+2*half+1]
    v2f av = *(const v2f*)(aptr + k0);
#pragma unroll
    for (int c = 0; c < 4; ++c) {
      int col  = (colgrp << 6) + (c << 4) + mr;        // 0..255
      int d    = col & (D_FEAT - 1);                   // W_emb output row
      int koff = (col < D_FEAT) ? 0 : (D_FEAT + 2);    // src cols vs dst cols
      // B 4x16 f32: v0 = B[2*half, n], v1 = B[2*half+1, n];  B[k,n] = We[d, koff+k]
      v2f bv = *(const v2f*)(We + (size_t)d * EDGE_IN + koff + k0 + 2 * half);
      acc[c] = __builtin_amdgcn_wmma_f32_16x16x4_f32(
          false, av, false, bv, (short)0, acc[c], false, false);
    }
  }
#pragma unroll
  for (int c = 0; c < 4; ++c) {
    int col = (colgrp << 6) + (c << 4) + mr;
#pragma unroll
    for (int r = 0; r < 8; ++r) {
      int row = m0 + r + 8 * half;      // C/D layout: lane-half selects M / M+8
      if (row < N) P[(size_t)row * PCOLS + col] = acc[c][r];
    }
  }
}

// ---- per-actor attention scalars: a_s = feat . Wa[0:128], a_d = feat . Wa[130:258]
__global__ __launch_bounds__(256) void a2a_attdot_kernel(
    const float* __restrict__ feat, const float* __restrict__ Wa,
    float* __restrict__ a, int N) {
  int wid  = (blockIdx.x * blockDim.x + threadIdx.x) >> 5;   // one wave per actor
  int lane = threadIdx.x & 31;
  if (wid >= N) return;
  float4 f  = *(const float4*)(feat + (size_t)wid * D_FEAT + lane * 4);
  float4 w  = *(const float4*)(Wa + lane * 4);
  const float* wd = Wa + D_FEAT + 2 + lane * 4;   // offset 130: not 16B aligned
  float ss = f.x * w.x + f.y * w.y + f.z * w.z + f.w * w.w;
  float sd = f.x * wd[0] + f.y * wd[1] + f.z * wd[2] + f.w * wd[3];
  for (int o = 16; o >= 1; o >>= 1) {
    ss += __shfl_xor(ss, o, 32);
    sd += __shfl_xor(sd, o, 32);
  }
  if (lane == 0) { a[2 * wid] = ss; a[2 * wid + 1] = sd; }
}

// ---- edge scores + segment max -------------------------------------------
__global__ void a2a_score_kernel(const float* __restrict__ a, const float* __restrict__ rel,
                                 const int* __restrict__ src, const int* __restrict__ dst,
                                 const float* __restrict__ Wa, float* __restrict__ sE,
                                 unsigned* __restrict__ maxu, int E) {
  int e = blockIdx.x * blockDim.x + threadIdx.x;
  if (e >= E) return;
  int si = src[e], di = dst[e];
  float s = a[2 * si] + a[2 * di + 1]
          + rel[2 * e] * Wa[D_FEAT] + rel[2 * e + 1] * Wa[D_FEAT + 1];
  s = (s > 0.f) ? s : 0.2f * s;          // LeakyReLU(0.2)
  sE[e] = s;
  atomicMax(&maxu[di], enc_f32(s));
}

// ---- exp(s - segmax) + segment sum ---------------------------------------
__global__ void a2a_exp_kernel(const int* __restrict__ dst, const unsigned* __restrict__ maxu,
                               float* __restrict__ sE, float* __restrict__ denom, int E) {
  int e = blockIdx.x * blockDim.x + threadIdx.x;
  if (e >= E) return;
  int di = dst[e];
  float ex = expf(sE[e] - dec_f32(maxu[di]));
  sE[e] = ex;
  atomAddF(&denom[di], ex);
}

// ---- weighted aggregation: one wave per edge, 128 f32 atomic adds ---------
__global__ __launch_bounds__(256) void a2a_agg_kernel(
    const float* __restrict__ P, const float* __restrict__ sE,
    const float* __restrict__ denom, const float* __restrict__ rel,
    const int* __restrict__ src, const int* __restrict__ dst,
    const float* __restrict__ wrel, float* __restrict__ agg, int E) {
  int e    = blockIdx.x * 8 + (threadIdx.x >> 5);
  int lane = threadIdx.x & 31;
  if (e >= E) return;
  int si = src[e], di = dst[e];
  float att = sE[e] / denom[di];
  float rx = rel[2 * e], ry = rel[2 * e + 1];
  float4 ps = *(const float4*)(P + (size_t)si * PCOLS + lane * 4);
  float4 pd = *(const float4*)(P + (size_t)di * PCOLS + D_FEAT + lane * 4);
  float4 w0 = *(const float4*)(wrel + lane * 4);
  float4 w1 = *(const float4*)(wrel + D_FEAT + lane * 4);
  float4 v;
  v.x = att * (ps.x + pd.x + rx * w0.x + ry * w1.x);
  v.y = att * (ps.y + pd.y + rx * w0.y + ry * w1.y);
  v.z = att * (ps.z + pd.z + rx * w0.z + ry * w1.z);
  v.w = att * (ps.w + pd.w + rx * w0.w + ry * w1.w);
  float* ap = agg + (size_t)di * D_FEAT + lane * 4;
  atomAddF(ap + 0, v.x);
  atomAddF(ap + 1, v.y);
  atomAddF(ap + 2, v.z);
  atomAddF(ap + 3, v.w);
}

// ---- ReLU ----------------------------------------------------------------
__global__ void a2a_relu_kernel(const float* __restrict__ agg, float* __restrict__ out, int n) {
  int i = blockIdx.x * blockDim.x + threadIdx.x;
  if (i < n) out[i] = fmaxf(agg[i], 0.f);
}

extern "C" void kernel_launch(void* const* d_in, const int* in_sizes, int n_in,
                              void* d_out, int out_size, void* d_ws, size_t ws_size,
                              hipStream_t stream) {
  const float* actor = (const float*)d_in[0];
  const float* rel   = (const float*)d_in[1];
  const float* W_att = (const float*)d_in[2];   // [L, 258]
  const float* W_emb = (const float*)d_in[3];   // [L, 128, 258]
  const int*   src   = (const int*)d_in[4];
  const int*   dst   = (const int*)d_in[5];
  int N = in_sizes[0] / D_FEAT;
  int E = in_sizes[4];
  int L = in_sizes[2] / EDGE_IN;
  float* out = (float*)d_out;

  // workspace layout (floats): P | a | sE | denom | maxu | wrel | agg | fbuf
  float*    P     = (float*)d_ws;
  float*    a     = P + (size_t)N * PCOLS;
  float*    sE    = a + (size_t)N * 2;
  float*    denom = sE + (size_t)E;
  unsigned* maxu  = (unsigned*)(denom + N);
  float*    wrel  = (float*)(maxu + N);
  float*    agg   = wrel + PCOLS;
  float*    fbuf  = agg + (size_t)N * D_FEAT;

  const float* fin = actor;
  int rowtiles  = (N + 15) / 16;
  int wavesTot  = rowtiles * 4;

  for (int l = 0; l < L; ++l) {
    const float* We = W_emb + (size_t)l * D_FEAT * EDGE_IN;
    const float* Wa = W_att + (size_t)l * EDGE_IN;
    float* fout = (l == L - 1) ? out : fbuf;

    a2a_init_kernel<<<(N * D_FEAT + 255) / 256, 256, 0, stream>>>(agg, denom, maxu, N);
    a2a_wrel_kernel<<<1, D_FEAT, 0, stream>>>(We, wrel);
    a2a_proj_wmma_kernel<<<(wavesTot + 7) / 8, 256, 0, stream>>>(fin, We, P, N);
    a2a_attdot_kernel<<<(N * 32 + 255) / 256, 256, 0, stream>>>(fin, Wa, a, N);
    a2a_score_kernel<<<(E + 255) / 256, 256, 0, stream>>>(a, rel, src, dst, Wa, sE, maxu, E);
    a2a_exp_kernel<<<(E + 255) / 256, 256, 0, stream>>>(dst, maxu, sE, denom, E);
    a2a_agg_kernel<<<(E + 7) / 8, 256, 0, stream>>>(P, sE, denom, rel, src, dst, wrel, agg, E);
    a2a_relu_kernel<<<(N * D_FEAT + 255) / 256, 256, 0, stream>>>(agg, fout, N * D_FEAT);

    fin = fbuf;
  }
}